// FRIENDSHIP_67680094650643
// MI455X (gfx1250) — compile-verified
//
#include <hip/hip_runtime.h>
#include <hip/hip_bf16.h>

// B=1024 users, E=32768 edges, C=64, D=64, FR=64, TAU=1e6, BIAS=1
// Bandwidth-bound: 512MB single-use stream of common_x -> non-temporal +
// async global->LDS staging (ASYNCcnt double-buffer) feeding f32 WMMA tiles.
// Math reassociated: score[e,c] = common_x[e,c,:] . (M @ [self_u; friend_e]),
// M = W_beta^T W_friend (64x128), precomputed once.

#define B_USERS 1024
#define E_EDGES 32768
#define NW 8
#define TPB (NW * 32)

typedef __attribute__((ext_vector_type(2))) float v2f;
typedef __attribute__((ext_vector_type(8))) float v8f;
typedef __attribute__((ext_vector_type(2))) int   v2i;

// -------- ws layout --------
// [0    .. 4100)  offsets int[1025]
// [4608 .. 37376) M float[64*128]

__global__ void scan_counts_kernel(const long long* __restrict__ counts,
                                   int* __restrict__ offsets) {
  __shared__ int s[B_USERS];
  int t = threadIdx.x;
  s[t] = (int)counts[t];
  __syncthreads();
  for (int o = 1; o < B_USERS; o <<= 1) {
    int add = (t >= o) ? s[t - o] : 0;
    __syncthreads();
    s[t] += add;
    __syncthreads();
  }
  offsets[t + 1] = s[t];
  if (t == 0) offsets[0] = 0;
}

// M[j,k] = sum_f W_beta[f,j] * W_friend[f,k]   (j<64, k<128)
__global__ void build_M_kernel(const float* __restrict__ Wf,   // [64,128]
                               const float* __restrict__ Wb,   // [64,64]
                               float* __restrict__ M) {
  int idx = blockIdx.x * blockDim.x + threadIdx.x;  // 8192 total
  int j = idx >> 7, k = idx & 127;
  float acc = 0.f;
  for (int f = 0; f < 64; ++f)
    acc = __builtin_fmaf(Wb[f * 64 + j], Wf[f * 128 + k], acc);
  M[idx] = acc;
}

__global__ __launch_bounds__(TPB)
void friendship_edge_kernel(const float* __restrict__ self_x,
                            const float* __restrict__ common_x,
                            const float* __restrict__ common_time,
                            const int*   __restrict__ common_mask,
                            const float* __restrict__ friend_x,
                            const int*   __restrict__ offsets,
                            const float* __restrict__ M,
                            float* __restrict__ out,
                            int maxN) {
  __shared__ float sM[64 * 129];            // pad 129: bank = (j + k) % 64
  __shared__ float sSelf[64];
  __shared__ float sFriend[NW][64];
  __shared__ float sV[NW][64];
  __shared__ float sTM[NW][64];             // exp(1 - t/tau) * mask, per edge
  __shared__ float sTile[NW][2][16 * 68];   // pad 68: bank = (4*row + k) % 64

  const int u     = blockIdx.x;
  const int tid   = threadIdx.x;
  const int lane  = tid & 31;
  const int wid   = tid >> 5;
  const int half  = lane >> 4;              // K-half select (ISA f32 A/B layout)
  const int row16 = lane & 15;

  // zero output row (padding columns must be exactly 0)
  for (int t = tid; t < maxN; t += TPB) out[(size_t)u * maxN + t] = 0.f;
  for (int i = tid; i < 64 * 128; i += TPB) {
    int j = i >> 7, k = i & 127;
    sM[j * 129 + k] = M[i];
  }
  if (tid < 64) sSelf[tid] = self_x[u * 64 + tid];
  __syncthreads();

  const int off = offsets[u];
  const int nE  = offsets[u + 1] - off;

  // async global->LDS stage of one 16x64 f32 tile (8 x b128 per lane),
  // landing directly in the bank-padded layout; non-temporal (single-use).
  auto issue_tile = [&](const float* cx, int t, int bufidx) {
    const float* tbase = cx + t * 1024;
    unsigned ldsBase = (unsigned)(size_t)(&sTile[wid][bufidx][0]);
#pragma unroll
    for (int q = 0; q < 8; ++q) {
      int p  = q * 32 + lane;               // float4 index 0..255
      int r  = p >> 4;
      int c4 = (p & 15) << 2;
      unsigned laddr = ldsBase + (unsigned)((r * 68 + c4) * 4);
      unsigned long long gaddr = (unsigned long long)(size_t)(tbase + p * 4);
      asm volatile("global_load_async_to_lds_b128 %0, %1, off th:TH_LOAD_NT"
                   :: "v"(laddr), "v"(gaddr) : "memory");
    }
  };

  // 16-step K march of V_WMMA_F32_16X16X4_F32 over one staged tile + epilogue
  auto compute_tile = [&](int e, int t, int bufidx, float& edgeAcc) {
    const float* tile = &sTile[wid][bufidx][0];
    v8f acc = {};
#pragma unroll
    for (int ks = 0; ks < 16; ++ks) {
      const int koff = ks * 4 + half * 2;   // lanes 0-15: K0,K1 ; 16-31: K2,K3
      v2f a, b;
      a[0] = tile[row16 * 68 + koff];
      a[1] = tile[row16 * 68 + koff + 1];
      b[0] = sV[wid][koff];
      b[1] = sV[wid][koff + 1];
      acc = __builtin_amdgcn_wmma_f32_16x16x4_f32(
          false, a, false, b, (short)0, acc, false, false);
    }
    const int cbase = t * 16 + half * 8;    // lane half picks row group of D
#pragma unroll
    for (int r = 0; r < 8; ++r) {
      float sc = acc[r];
      float sp = fmaxf(sc, 0.f) + log1pf(__expf(-fabsf(sc)));  // stable softplus
      edgeAcc += sp * sTM[wid][cbase + r];
    }
  };

  for (int ei = wid; ei < nE; ei += NW) {
    const int e = off + ei;
    const float* cx = common_x + (size_t)e * 4096;
    if (ei + NW < nE)
      __builtin_prefetch(common_x + (size_t)(e + NW) * 4096, 0, 1);

    // start streaming tiles 0,1 before the matvec so DRAM latency overlaps
    issue_tile(cx, 0, 0);
    issue_tile(cx, 1, 1);

    { // friend row (non-temporal, single-use)
      v2f fv = __builtin_nontemporal_load(
          ((const v2f*)(friend_x + (size_t)e * 64)) + lane);
      sFriend[wid][2 * lane]     = fv[0];
      sFriend[wid][2 * lane + 1] = fv[1];
    }
    { // temporal*mask weights for this edge, computed once per c
      v2f tv = __builtin_nontemporal_load(
          ((const v2f*)(common_time + (size_t)e * 64)) + lane);
      v2i mv = __builtin_nontemporal_load(
          ((const v2i*)(common_mask + (size_t)e * 64)) + lane);
      sTM[wid][2 * lane]     = __expf(1.0f - tv[0] * 1.0e-6f) * (float)mv[0];
      sTM[wid][2 * lane + 1] = __expf(1.0f - tv[1] * 1.0e-6f) * (float)mv[1];
    }

    // v = M @ [self ; friend] ; lane computes j = lane and j = lane+32
    float v0 = 0.f, v1 = 0.f;
    const int j0 = lane, j1 = lane + 32;
    for (int k = 0; k < 64; ++k) {
      float xs = sSelf[k];
      v0 = __builtin_fmaf(sM[j0 * 129 + k], xs, v0);
      v1 = __builtin_fmaf(sM[j1 * 129 + k], xs, v1);
    }
    for (int k = 0; k < 64; ++k) {
      float xf = sFriend[wid][k];
      v0 = __builtin_fmaf(sM[j0 * 129 + 64 + k], xf, v0);
      v1 = __builtin_fmaf(sM[j1 * 129 + 64 + k], xf, v1);
    }
    sV[wid][lane]      = v0;
    sV[wid][lane + 32] = v1;

    float edgeAcc = 0.f;
    // double-buffered pipeline: wait for the older 8 async ops, compute,
    // immediately refill the consumed buffer with the tile after next.
    asm volatile("s_wait_asynccnt 0x8" ::: "memory");
    compute_tile(e, 0, 0, edgeAcc);
    issue_tile(cx, 2, 0);
    asm volatile("s_wait_asynccnt 0x8" ::: "memory");
    compute_tile(e, 1, 1, edgeAcc);
    issue_tile(cx, 3, 1);
    asm volatile("s_wait_asynccnt 0x8" ::: "memory");
    compute_tile(e, 2, 0, edgeAcc);
    asm volatile("s_wait_asynccnt 0x0" ::: "memory");
    compute_tile(e, 3, 1, edgeAcc);

    // columns of D are duplicates; rows 0-7 live in lane 0, rows 8-15 in lane 16
    float r0 = __shfl(edgeAcc, 0, 32);
    float r1 = __shfl(edgeAcc, 16, 32);
    if (lane == 0) out[(size_t)u * maxN + ei] = r0 + r1;
  }
}

extern "C" void kernel_launch(void* const* d_in, const int* in_sizes, int n_in,
                              void* d_out, int out_size, void* d_ws, size_t ws_size,
                              hipStream_t stream) {
  const float*     self_x   = (const float*)d_in[0];
  const float*     common_x = (const float*)d_in[1];
  const float*     ctime    = (const float*)d_in[2];
  const int*       cmask    = (const int*)d_in[3];
  const float*     friend_x = (const float*)d_in[4];
  const long long* counts   = (const long long*)d_in[5];   // np.int64
  const float*     Wf       = (const float*)d_in[6];       // [64,128]
  const float*     Wb       = (const float*)d_in[7];       // [64,64]

  int*   offsets = (int*)d_ws;
  float* M       = (float*)((char*)d_ws + 4608);
  float* out     = (float*)d_out;
  int maxN = out_size / B_USERS;

  scan_counts_kernel<<<1, B_USERS, 0, stream>>>(counts, offsets);
  build_M_kernel<<<32, 256, 0, stream>>>(Wf, Wb, M);
  friendship_edge_kernel<<<B_USERS, TPB, 0, stream>>>(
      self_x, common_x, ctime, cmask, friend_x, offsets, M, out, maxN);
}